// FCL3layers_36653250904914
// MI455X (gfx1250) — compile-verified
//
#include <hip/hip_runtime.h>
#include <hip/hip_bf16.h>

typedef __attribute__((ext_vector_type(2))) float v2f;
typedef __attribute__((ext_vector_type(8))) float v8f;

#define DIM_D   320
#define DIM_D1  10
#define DIM_D2  5
#define TILE_M  16
#define WAVES_PER_BLOCK 8

// Kernel 1: per-wave 16-row tile of  t = relu( relu(x @ W1^T) @ W2^T + b2 )
// using V_WMMA_F32_16X16X4_F32 for the dominant 320x10 GEMM.
__global__ __launch_bounds__(256) void fcl3_gemm_kernel(
    const float* __restrict__ x,   // [N, 320]
    const float* __restrict__ W1,  // [10, 320]
    const float* __restrict__ W2,  // [5, 10]
    const float* __restrict__ b2,  // [5]
    float* __restrict__ t,         // [N, 5] (workspace)
    int nrows)
{
    // B matrix (W1^T padded to 16 cols): sB[k][n], n in [0,16)
    __shared__ float sB[DIM_D * 16];                       // 20 KB
    __shared__ float sH[WAVES_PER_BLOCK][TILE_M][16];      // 8 KB

    const int tid = threadIdx.x;

    // Stage W1^T into LDS, zero-padding columns 10..15 (coalesced fill).
    for (int idx = tid; idx < DIM_D * 16; idx += blockDim.x) {
        const int k = idx >> 4;
        const int n = idx & 15;
        sB[idx] = (n < DIM_D1) ? W1[n * DIM_D + k] : 0.0f;
    }
    __syncthreads();

    const int wave = tid >> 5;
    const int lane = tid & 31;
    const int m    = lane & 15;        // row within tile (A) / col (B,C)
    const int koff = (lane >> 4) << 1; // 0 for lanes 0-15, 2 for lanes 16-31

    const int numTiles = (nrows + TILE_M - 1) / TILE_M;
    int tile = blockIdx.x * WAVES_PER_BLOCK + wave;
    const bool active = (tile < numTiles);
    if (!active) tile = 0;   // keep wave alive (uniform) so EXEC==all-1s at WMMA

    const float* xrow = x + (size_t)(tile * TILE_M + m) * DIM_D;

    v8f c = {0.f, 0.f, 0.f, 0.f, 0.f, 0.f, 0.f, 0.f};

    #pragma unroll 8
    for (int k0 = 0; k0 < DIM_D; k0 += 4) {
        // A fragment: 16x4 f32, ISA layout (lanes 0-15: K=k0,k0+1; 16-31: K=k0+2,k0+3).
        // x is streamed exactly once -> non-temporal load (don't pollute L2).
        v2f a = __builtin_nontemporal_load((const v2f*)(xrow + k0 + koff));
        // B fragment: 4x16 f32 from LDS, same K split, N = m
        v2f b;
        b.x = sB[(k0 + koff)     * 16 + m];
        b.y = sB[(k0 + koff + 1) * 16 + m];
        c = __builtin_amdgcn_wmma_f32_16x16x4_f32(
                /*neg_a=*/false, a, /*neg_b=*/false, b,
                /*c_mod=*/(short)0, c, /*reuse_a=*/false, /*reuse_b=*/false);
    }

    // ReLU and spill the 16x16 H tile to LDS.
    // C layout: lane l holds C[((l>>4)*8)+vg][l&15] in c[vg].
    #pragma unroll
    for (int vg = 0; vg < 8; ++vg) {
        const int mr = ((lane >> 4) << 3) + vg;
        const float hv = c[vg];
        sH[wave][mr][m] = hv > 0.0f ? hv : 0.0f;
    }
    __syncthreads();

    // Second (tiny) layer: lanes 0..15 each own one row of the tile.
    if (active && lane < 16) {
        const int row = tile * TILE_M + m;
        if (row < nrows) {
            float h[DIM_D1];
            #pragma unroll
            for (int i = 0; i < DIM_D1; ++i) h[i] = sH[wave][m][i];
            #pragma unroll
            for (int j = 0; j < DIM_D2; ++j) {
                float acc = b2[j];
                #pragma unroll
                for (int i = 0; i < DIM_D1; ++i)
                    acc += h[i] * W2[j * DIM_D1 + i];
                // t is re-read by kernel 2: keep regular-temporal (L2 resident).
                t[(size_t)row * DIM_D2 + j] = acc > 0.0f ? acc : 0.0f;
            }
        }
    }
}

// Kernel 2: zero-padded window of 5 rows over t[N,5] -> out[N,25].
// Because rows of t are contiguous, out[n*25 + i] == t[(n-2)*5 + i] (shifted copy).
__global__ __launch_bounds__(256) void fcl3_window_kernel(
    const float* __restrict__ t, float* __restrict__ out, int nrows)
{
    const int total = nrows * 25;
    int idx = blockIdx.x * blockDim.x + threadIdx.x;
    if (idx >= total) return;
    const int n = idx / 25;
    const int i = idx - n * 25;
    const int s = (n - 2) * 5 + i;           // flat source index into t
    const float v = (s >= 0 && s < nrows * 5) ? t[s] : 0.0f;
    // out is write-once -> non-temporal store.
    __builtin_nontemporal_store(v, out + idx);
}

extern "C" void kernel_launch(void* const* d_in, const int* in_sizes, int n_in,
                              void* d_out, int out_size, void* d_ws, size_t ws_size,
                              hipStream_t stream) {
    const float* x  = (const float*)d_in[0];   // [N, 320]
    const float* W1 = (const float*)d_in[1];   // [10, 320]
    const float* W2 = (const float*)d_in[2];   // [5, 10]
    const float* b2 = (const float*)d_in[3];   // [5]
    float* out = (float*)d_out;                // [N, 25]
    float* t   = (float*)d_ws;                 // [N, 5] scratch

    const int nrows = in_sizes[0] / DIM_D;     // 200000

    const int numTiles = (nrows + TILE_M - 1) / TILE_M;
    const int blocks1  = (numTiles + WAVES_PER_BLOCK - 1) / WAVES_PER_BLOCK;
    fcl3_gemm_kernel<<<blocks1, 256, 0, stream>>>(x, W1, W2, b2, t, nrows);

    const int total   = nrows * 25;
    const int blocks2 = (total + 255) / 256;
    fcl3_window_kernel<<<blocks2, 256, 0, stream>>>(t, out, nrows);
}